// Quantum_Evo_Layer_58042188038683
// MI455X (gfx1250) — compile-verified
//
#include <hip/hip_runtime.h>
#include <hip/hip_bf16.h>

// Quantum evolution layer: out[bt] = distributions of expm(-i*L*H[bt]) @ psi0
// B*T = 2048 matrices of 64x64. Scaling-and-squaring (s=5) + Horner Taylor (m=8),
// all matmuls via v_wmma_f32_16x16x4_f32 on 16x16 tiles held in LDS.

typedef float v2f __attribute__((ext_vector_type(2)));
typedef float v8f __attribute__((ext_vector_type(8)));

#define NDIM   64
#define LDA    66            // padded LDS row stride (floats)
#define MSIZE  (NDIM * LDA)
#define BLOCK  512           // 16 waves (wave32): one 16x16 output tile per wave

__device__ __forceinline__ v8f wmma_f32(v2f a, v2f b, v8f c) {
  // (neg_a, A, neg_b, B, c_mod, C, reuse_a, reuse_b)
  return __builtin_amdgcn_wmma_f32_16x16x4_f32(false, a, false, b, (short)0, c,
                                               false, false);
}

// A-operand fragment: 16x4 tile of left matrix. lane%16 = row within tile,
// lane/16 selects K pair {0,1} vs {2,3}. (ISA 7.12.2, 32-bit A 16x4 layout)
__device__ __forceinline__ v2f load_afrag(const float* Lm, int tr, int k4,
                                          int mn, int half) {
  const float* p = Lm + (tr * 16 + mn) * LDA + (k4 + half * 2);
  v2f a;
  a.x = p[0];
  a.y = p[1];
  return a;
}

// B-operand fragment: 4x16 tile of right matrix. lane%16 = col within tile,
// lane/16 selects K pair.
__device__ __forceinline__ v2f load_bfrag(const float* Rm, int tc, int k4,
                                          int mn, int half) {
  int col = tc * 16 + mn;
  int kk = k4 + half * 2;
  v2f b;
  b.x = Rm[kk * LDA + col];
  b.y = Rm[(kk + 1) * LDA + col];
  return b;
}

__global__ __launch_bounds__(BLOCK)
void quantum_evo_expm_kernel(const float* __restrict__ x,
                             const float* __restrict__ psi0,
                             float* __restrict__ out) {
  __shared__ float sAi[MSIZE];      // imag part of scaled generator: -c*H
  __shared__ float sBuf[4][MSIZE];  // ping-pong complex buffers (r,i)x2

  const int bt = blockIdx.x;
  const float c = 1.0f / 32.0f;     // LENGTH / 2^s, LENGTH=1, s=5

  const int tid  = threadIdx.x;
  const int lane = tid & 31;
  const int wave = tid >> 5;
  const int half = lane >> 4;       // 0: lanes 0-15, 1: lanes 16-31
  const int mn   = lane & 15;
  const int tr   = wave >> 2;       // output tile row (0..3)
  const int tc   = wave & 3;        // output tile col (0..3)

  // ---- Load H; init Ai = -c*H, Rr = I, Ri = Ai/8 (Horner innermost I + A/8)
  const float* Hg = x + (size_t)bt * (NDIM * NDIM);
  for (int i = tid; i < (NDIM * NDIM) / 4; i += BLOCK) {
    int idx = i * 4;
    int row = idx >> 6;
    int col = idx & 63;
    float4 h = *reinterpret_cast<const float4*>(Hg + idx);
    float a0 = -c * h.x, a1 = -c * h.y, a2 = -c * h.z, a3 = -c * h.w;
    float* pa  = sAi      + row * LDA + col;
    float* prr = sBuf[0]  + row * LDA + col;
    float* pri = sBuf[1]  + row * LDA + col;
    pa[0] = a0; pa[1] = a1; pa[2] = a2; pa[3] = a3;
    prr[0] = (row == col + 0) ? 1.0f : 0.0f;
    prr[1] = (row == col + 1) ? 1.0f : 0.0f;
    prr[2] = (row == col + 2) ? 1.0f : 0.0f;
    prr[3] = (row == col + 3) ? 1.0f : 0.0f;
    pri[0] = a0 * 0.125f; pri[1] = a1 * 0.125f;
    pri[2] = a2 * 0.125f; pri[3] = a3 * 0.125f;
  }

  float* Rr = sBuf[0];
  float* Ri = sBuf[1];
  float* Tr = sBuf[2];
  float* Ti = sBuf[3];
  __syncthreads();

  // ---- Horner steps j = 7..1:  R <- I + (A/j)*R, A purely imaginary:
  //      newRr = I - (1/j)*(Ai*Ri),  newRi = (1/j)*(Ai*Rr)
  for (int j = 7; j >= 1; --j) {
    const float inv = 1.0f / (float)j;
    v8f P = {};  // Ai * Rr
    v8f Q = {};  // Ai * Ri
#pragma unroll
    for (int k4 = 0; k4 < NDIM; k4 += 4) {
      v2f a  = load_afrag(sAi, tr, k4, mn, half);
      v2f br = load_bfrag(Rr,  tc, k4, mn, half);
      v2f bi = load_bfrag(Ri,  tc, k4, mn, half);
      P = wmma_f32(a, br, P);
      Q = wmma_f32(a, bi, Q);
    }
#pragma unroll
    for (int v = 0; v < 8; ++v) {
      int row = tr * 16 + v + half * 8;
      int col = tc * 16 + mn;
      Tr[row * LDA + col] = ((row == col) ? 1.0f : 0.0f) - inv * Q[v];
      Ti[row * LDA + col] = inv * P[v];
    }
    __syncthreads();
    float* t;
    t = Rr; Rr = Tr; Tr = t;
    t = Ri; Ri = Ti; Ti = t;
  }

  // ---- 5 squarings: R <- R*R (full complex: 4 real matmuls)
  for (int sq = 0; sq < 5; ++sq) {
    v8f Crr = {}, Cii = {}, Cri = {}, Cir = {};
#pragma unroll
    for (int k4 = 0; k4 < NDIM; k4 += 4) {
      v2f ar = load_afrag(Rr, tr, k4, mn, half);
      v2f ai = load_afrag(Ri, tr, k4, mn, half);
      v2f br = load_bfrag(Rr, tc, k4, mn, half);
      v2f bi = load_bfrag(Ri, tc, k4, mn, half);
      Crr = wmma_f32(ar, br, Crr);
      Cii = wmma_f32(ai, bi, Cii);
      Cri = wmma_f32(ar, bi, Cri);
      Cir = wmma_f32(ai, br, Cir);
    }
#pragma unroll
    for (int v = 0; v < 8; ++v) {
      int row = tr * 16 + v + half * 8;
      int col = tc * 16 + mn;
      Tr[row * LDA + col] = Crr[v] - Cii[v];
      Ti[row * LDA + col] = Cri[v] + Cir[v];
    }
    __syncthreads();
    float* t;
    t = Rr; Rr = Tr; Tr = t;
    t = Ri; Ri = Ti; Ti = t;
  }

  // ---- Epilogue: v = U @ psi0; power/interferometer distributions
  if (tid < NDIM) {
    const int n = tid;
    float vr = 0.0f, vi = 0.0f;
    for (int j = 0; j < NDIM; ++j) {
      float p = psi0[j];
      vr += Rr[n * LDA + j] * p;
      vi += Ri[n * LDA + j] * p;
    }
    float* o = out + (size_t)bt * (2 * NDIM);
    float pr = 1.0f + vr;
    float qi = 1.0f + vi;
    o[n]        = 0.25f * (pr * pr + vi * vi);  // |0.5*(1 + v)|^2
    o[NDIM + n] = 0.25f * (vr * vr + qi * qi);  // |0.5*(i + v)|^2
  }
}

extern "C" void kernel_launch(void* const* d_in, const int* in_sizes, int n_in,
                              void* d_out, int out_size, void* d_ws, size_t ws_size,
                              hipStream_t stream) {
  const float* x    = (const float*)d_in[0];   // [B,T,64,64] f32
  const float* psi0 = (const float*)d_in[1];   // [64,1] f32
  float* out = (float*)d_out;                  // [B,T,128] f32
  const int BT = in_sizes[0] / (NDIM * NDIM);  // 2048
  quantum_evo_expm_kernel<<<BT, BLOCK, 0, stream>>>(x, psi0, out);
}